// VectorQuantizer_27882927686136
// MI455X (gfx1250) — compile-verified
//
#include <hip/hip_runtime.h>

// ---------------- problem constants (fixed shapes from reference) ----------
#define C_DIM   256
#define NE      8192
#define NTOT    16384          // 16 * 32 * 32
#define MTILE   64             // z rows per block
#define KCHUNK  32             // codebook rows per staged chunk
#define NCHUNK  (NE / KCHUNK)  // 256
#define Z2S     80             // z-tile row stride in float2 units (Z2S*2 % 64 == 32 -> perfect b64 banks)
#define CTS     260            // ct row stride in dwords (260 % 64 == 4 -> perfect b64 banks)
#define CTBUF   (KCHUNK * CTS) // 8320 floats per ct buffer

typedef __attribute__((ext_vector_type(2))) float v2f;
typedef __attribute__((ext_vector_type(4))) float v4f;
typedef __attribute__((ext_vector_type(8))) float v8f;
typedef __attribute__((ext_vector_type(4))) int   v4i;

// smem floats: zt2 (128*Z2S*2 = 20480) + 2*CTBUF (16640) + redv 128 + redi 128 + fidx 64 + lsum 8
#define SMEM_FLOATS (128 * Z2S * 2 + 2 * CTBUF + 128 + 128 + 64 + 8)

// -------- CDNA5 async global->LDS path (guarded; falls back to sync copies) ----
#if defined(__has_builtin)
#  if __has_builtin(__builtin_amdgcn_global_load_async_to_lds_b128) && \
      __has_builtin(__builtin_amdgcn_s_wait_asynccnt)
#    define VQ_HAVE_ASYNC 1
#  endif
#endif

typedef __attribute__((address_space(1))) v4i gv4i_t;   // global int4
typedef __attribute__((address_space(3))) v4i sv4i_t;   // LDS int4
typedef __attribute__((address_space(1))) void gvoid_t;
typedef __attribute__((address_space(3))) void svoid_t;

__device__ __forceinline__ void cp_b128(const float* g, float* l) {
#ifdef VQ_HAVE_ASYNC
    __builtin_amdgcn_global_load_async_to_lds_b128(
        (gv4i_t*)(gvoid_t*)(void*)g,
        (sv4i_t*)(svoid_t*)(void*)l, 0, 0);
#else
    *(v4f*)l = *(const v4f*)g;
#endif
}

__device__ __forceinline__ void wait_async() {
#ifdef VQ_HAVE_ASYNC
    __builtin_amdgcn_s_wait_asynccnt(0);
#endif
}

// stage one KCHUNK x C_DIM codebook chunk row-major into LDS (stride CTS)
__device__ __forceinline__ void stage_ct(const float* __restrict__ codebook,
                                         float* dst, int k0, int tid) {
#pragma unroll
    for (int j = 0; j < (KCHUNK * C_DIM) / (4 * 256); ++j) {  // 8 x b128 per thread
        int f4 = j * 256 + tid;
        int k  = f4 >> 6;        // 64 float4 per codebook row
        int c4 = f4 & 63;
        cp_b128(codebook + (size_t)(k0 + k) * C_DIM + c4 * 4,
                dst + k * CTS + c4 * 4);
    }
}

// -------------------- kernel 1: codebook row norms -------------------------
__global__ void __launch_bounds__(256)
vq_norms(const float* __restrict__ codebook, float* __restrict__ norms) {
    const int lane = threadIdx.x & 31;
    const int wave = threadIdx.x >> 5;
    const int row  = blockIdx.x * 8 + wave;
    const float* cb = codebook + (size_t)row * C_DIM;
    float s = 0.f;
#pragma unroll
    for (int j = 0; j < C_DIM / 32; ++j) {
        float v = cb[j * 32 + lane];
        s += v * v;
    }
#pragma unroll
    for (int off = 16; off >= 1; off >>= 1) s += __shfl_xor(s, off, 32);
    if (lane == 0) norms[row] = s;
}

// -------------------- kernel 2: fused GEMM + argmin + gather ---------------
__global__ void __launch_bounds__(256)
vq_main(const float* __restrict__ z, const float* __restrict__ codebook,
        const float* __restrict__ norms, float* __restrict__ zq_out,
        int* __restrict__ idx_out, float* __restrict__ partial) {
    extern __shared__ float smem[];
    float* zt2f = smem;                       // [128 c-pairs][Z2S] float2 (as floats)
    float* ct0  = zt2f + 128 * Z2S * 2;       // [KCHUNK][CTS] buffer 0
    float* ct1  = ct0 + CTBUF;                // buffer 1
    float* redv = ct1 + CTBUF;                // [2][64]
    int*   redi = (int*)(redv + 128);         // [2][64]
    int*   fidx = redi + 128;                 // [64]
    float* lsum = (float*)(fidx + 64);        // [8]

    const int tid  = threadIdx.x;
    const int lane = tid & 31;
    const int wave = tid >> 5;
    const int l16  = lane & 15;
    const int hi   = lane >> 4;            // 0: lanes 0-15, 1: lanes 16-31
    const int n0   = blockIdx.x * MTILE;   // first z row of this block
    const int b    = n0 >> 10;             // batch (1024 spatial per batch)
    const int hw0  = n0 & 1023;

    const float* zb = z + (size_t)b * C_DIM * 1024 + hw0;

    // ---- stage Z tile as (c,c+1) float2 pairs: zt2[c2][m] ----
    for (int i = tid; i < 128 * MTILE; i += 256) {
        int c2 = i >> 6, m = i & 63;
        v2f p;
        p.x = zb[(2 * c2) * 1024 + m];
        p.y = zb[(2 * c2 + 1) * 1024 + m];
        *(v2f*)(zt2f + (c2 * Z2S + m) * 2) = p;
    }

    const int m0w = (wave & 3) * 16;   // wave's 16-row M subtile
    const int kk  = (wave >> 2) * 16;  // wave's 16-code K subtile within chunk

    // A fragment base: lane supplies Z[m][c0+2*hi], Z[m][c0+1+2*hi] as one b64
    const float* zAf = zt2f + (hi * Z2S + m0w + l16) * 2;

    float minv[8];
    int   mini[8];
#pragma unroll
    for (int r = 0; r < 8; ++r) { minv[r] = 3.4e38f; mini[r] = 0; }

    // prologue: kick off chunk 0 into buffer 0
    stage_ct(codebook, ct0, 0, tid);

    for (int kc = 0; kc < NCHUNK; ++kc) {
        float* cur = (kc & 1) ? ct1 : ct0;
        float* nxt = (kc & 1) ? ct0 : ct1;

        wait_async();        // own wave's async writes to `cur` complete
        __syncthreads();     // everyone's writes visible; prev compute of `nxt` done

        if (kc + 1 < NCHUNK)
            stage_ct(codebook, nxt, (kc + 1) * KCHUNK, tid);  // overlaps compute below

        // B fragment base: lane supplies E[code][c0+2*hi], E[code][c0+1+2*hi] as one b64
        const float* Bp = cur + (kk + l16) * CTS + 2 * hi;

        v8f acc = {};
#pragma unroll 8
        for (int c02 = 0; c02 < 128; c02 += 2) {          // c0 = 2*c02, 64 WMMAs
            v2f a  = *(const v2f*)(zAf + c02 * (2 * Z2S));
            v2f bm = *(const v2f*)(Bp + c02 * 2);
            acc = __builtin_amdgcn_wmma_f32_16x16x4_f32(
                false, a, false, bm, (short)0, acc, false, false);
        }

        // score = ||e||^2 - 2*dot ; ||z||^2 is row-constant, drops from argmin
        const int   code = kc * KCHUNK + kk + l16;
        const float nrm  = norms[code];
#pragma unroll
        for (int r = 0; r < 8; ++r) {
            float s = __builtin_fmaf(-2.0f, acc[r], nrm);
            if (s < minv[r]) { minv[r] = s; mini[r] = code; }  // first-min kept (codes ascend)
        }
    }

    // ---- argmin reduce over the 16 lanes holding one row's codes ----
    const int part = wave >> 2;
#pragma unroll
    for (int r = 0; r < 8; ++r) {
        float v = minv[r];
        int   ii = mini[r];
#pragma unroll
        for (int off = 1; off < 16; off <<= 1) {
            float ov = __shfl_xor(v, off, 32);
            int   oi = __shfl_xor(ii, off, 32);
            if (ov < v || (ov == v && oi < ii)) { v = ov; ii = oi; }
        }
        if (l16 == 0) {
            int row = m0w + r + hi * 8;   // C/D layout: VGPR r -> rows r / r+8
            redv[part * 64 + row] = v;
            redi[part * 64 + row] = ii;
        }
    }
    __syncthreads();

    if (tid < 64) {
        float v0 = redv[tid], v1 = redv[64 + tid];
        int   i0 = redi[tid], i1 = redi[64 + tid];
        int f = (v1 < v0 || (v1 == v0 && i1 < i0)) ? i1 : i0;
        fidx[tid] = f;
        idx_out[n0 + tid] = f;
    }
    __syncthreads();

    // ---- gather z_q (NCHW) + deterministic loss partial; z tile still in LDS ----
    float* outb = zq_out + (size_t)b * C_DIM * 1024 + hw0;
    float myloss = 0.f;
    for (int i = tid; i < C_DIM * MTILE; i += 256) {
        int c = i >> 6, m = i & 63;
        float q  = codebook[(size_t)fidx[m] * C_DIM + c];
        float zv = zt2f[((c >> 1) * Z2S + m) * 2 + (c & 1)];
        float d  = q - zv;
        myloss = __builtin_fmaf(d, d, myloss);
        outb[c * 1024 + m] = q;
    }
#pragma unroll
    for (int off = 16; off >= 1; off >>= 1) myloss += __shfl_xor(myloss, off, 32);
    if (lane == 0) lsum[wave] = myloss;
    __syncthreads();
    if (tid == 0) {
        float s = 0.f;
        for (int w = 0; w < 8; ++w) s += lsum[w];
        partial[blockIdx.x] = s;   // fixed-order sums -> bit-stable across graph replays
    }
}

// -------------------- kernel 3: finalize scalar loss -----------------------
__global__ void vq_loss(const float* __restrict__ partial, float* __restrict__ loss_out) {
    if (threadIdx.x == 0 && blockIdx.x == 0) {
        float s = 0.f;
        for (int i = 0; i < NTOT / MTILE; ++i) s += partial[i];
        // loss = mean((zq-z)^2) + 0.25*mean((zq-z)^2) = 1.25 * sum / (N*C)
        loss_out[0] = 1.25f * s / 4194304.0f;
    }
}

// ---------------------------------------------------------------------------
extern "C" void kernel_launch(void* const* d_in, const int* in_sizes, int n_in,
                              void* d_out, int out_size, void* d_ws, size_t ws_size,
                              hipStream_t stream) {
    const float* zin      = (const float*)d_in[0];   // [16,256,32,32]
    const float* codebook = (const float*)d_in[1];   // [8192,256]

    float* out   = (float*)d_out;
    float* zq    = out;                    // 4194304 f32
    float* loss  = out + 4194304;          // 1 f32
    int*   idx   = (int*)(out + 4194305);  // 16384 i32 (bit-stored)

    float* norms   = (float*)d_ws;         // 8192 f32
    float* partial = norms + NE;           // 256 f32

    vq_norms<<<NE / 8, 256, 0, stream>>>(codebook, norms);

    const size_t shmem = SMEM_FLOATS * sizeof(float);  // ~150 KB (<= 320 KB/WG per ISA)
    vq_main<<<NTOT / MTILE, 256, shmem, stream>>>(zin, codebook, norms, zq, idx, partial);

    vq_loss<<<1, 64, 0, stream>>>(partial, loss);
}